// RobustSmoothLoss_31086973289268
// MI455X (gfx1250) — compile-verified
//
#include <hip/hip_runtime.h>
#include <hip/hip_bf16.h>
#include <math.h>

typedef __attribute__((ext_vector_type(2))) float v2f;
typedef __attribute__((ext_vector_type(8))) float v8f;

#define CDIM   4096
#define NROWS  (128 * 64)

// hyperparameters from the reference
#define NUM_ITER_F 100000.0f
#define TRANSIT    0.2f
#define EXP_BASE_F 1.0f
#define ALPHA      0.1f
#define LN_C       8.31776616672f   /* ln(4096) */

// Sum the 16x16 f32 WMMA accumulator to a single wave-wide scalar.
// With B == all-ones, every column of D is identical; per-lane sum of the
// 8 C VGPRs covers rows {0..7} (lanes 0-15) / {8..15} (lanes 16-31), so one
// xor-16 shuffle completes the 16-row sum (same value in every lane).
__device__ __forceinline__ float wmma_acc_total(v8f c) {
    float s = c[0] + c[1] + c[2] + c[3] + c[4] + c[5] + c[6] + c[7];
    s += __shfl_xor(s, 16, 32);
    return s;
}

__global__ __launch_bounds__(256)
void row_stats_kernel(const float* __restrict__ input,
                      const int*   __restrict__ target,
                      const int*   __restrict__ cur_time,
                      float*       __restrict__ ws)
{
    __shared__ __align__(16) float row[CDIM];   // 16 KB staged row
    __shared__ float redS[8];
    __shared__ float redT[8];
    __shared__ float bcast[1];

    const int r    = blockIdx.x;
    const int tid  = threadIdx.x;
    const int lane = tid & 31;
    const int wave = tid >> 5;

    const float* gsrc = input + (size_t)r * CDIM;

    // ---- stage row into LDS via CDNA5 async global->LDS DMA path ----
    {
        unsigned lds_base = (unsigned)(size_t)(&row[0]);  // low 32b = LDS offset
        #pragma unroll
        for (int i = 0; i < 4; ++i) {
            unsigned byte_off = (unsigned)(i * 1024 + tid * 4) * 4u;
            unsigned lds_addr = lds_base + byte_off;
            asm volatile("global_load_async_to_lds_b128 %0, %1, %2"
                         :: "v"(lds_addr), "v"(byte_off), "s"(gsrc)
                         : "memory");
        }
        asm volatile("s_wait_asynccnt 0" ::: "memory");
    }
    __syncthreads();

    const float4* rowv = (const float4*)row;

    // ---- pass 1: row max (LDS-resident, no extra HBM traffic) ----
    float m = -3.402823466e38f;
    #pragma unroll
    for (int i = 0; i < 4; ++i) {
        float4 x = rowv[i * 256 + tid];
        m = fmaxf(m, fmaxf(fmaxf(x.x, x.y), fmaxf(x.z, x.w)));
    }
    #pragma unroll
    for (int off = 16; off; off >>= 1)
        m = fmaxf(m, __shfl_xor(m, off, 32));
    if (lane == 0) redS[wave] = m;
    __syncthreads();
    if (tid == 0) {
        float mm = redS[0];
        #pragma unroll
        for (int w = 1; w < 8; ++w) mm = fmaxf(mm, redS[w]);
        bcast[0] = mm;
    }
    __syncthreads();
    m = bcast[0];

    // ---- pass 2: S = sum exp(x-m), T = sum x*exp(x-m) ----
    // exp on VALU; additions absorbed by the matrix pipe (co-executing
    // v_wmma_f32_16x16x4_f32 with all-ones B).
    v8f cs = {0.f, 0.f, 0.f, 0.f, 0.f, 0.f, 0.f, 0.f};
    v8f ct = {0.f, 0.f, 0.f, 0.f, 0.f, 0.f, 0.f, 0.f};
    const v2f one2 = {1.0f, 1.0f};
    #pragma unroll
    for (int i = 0; i < 4; ++i) {
        float4 x = rowv[i * 256 + tid];
        float e0 = __expf(x.x - m), e1 = __expf(x.y - m);
        float e2 = __expf(x.z - m), e3 = __expf(x.w - m);
        v2f a0 = {e0, e1},           a1 = {e2, e3};
        v2f t0 = {x.x * e0, x.y * e1}, t1 = {x.z * e2, x.w * e3};
        cs = __builtin_amdgcn_wmma_f32_16x16x4_f32(false, a0, false, one2, (short)0, cs, false, false);
        cs = __builtin_amdgcn_wmma_f32_16x16x4_f32(false, a1, false, one2, (short)0, cs, false, false);
        ct = __builtin_amdgcn_wmma_f32_16x16x4_f32(false, t0, false, one2, (short)0, ct, false, false);
        ct = __builtin_amdgcn_wmma_f32_16x16x4_f32(false, t1, false, one2, (short)0, ct, false, false);
    }
    float s_wave = wmma_acc_total(cs);
    float t_wave = wmma_acc_total(ct);

    __syncthreads();   // redS reuse: everyone is past the max-phase reads
    if (lane == 0) { redS[wave] = s_wave; redT[wave] = t_wave; }
    __syncthreads();

    // ---- per-row epilogue (thread 0) ----
    if (tid == 0) {
        float S = 0.f, T = 0.f;
        #pragma unroll
        for (int w = 0; w < 8; ++w) { S += redS[w]; T += redT[w]; }

        float logZ = m + __logf(S);
        float H    = logZ - T / S;                  // entropy of softmax

        int   tg    = target[r];
        bool  valid = (tg != 0);                    // ignore_index == 0
        float xt    = row[valid ? tg : 0];
        float nll   = logZ - xt;
        float pt    = __expf(xt - logZ);

        float tr  = (float)cur_time[0] * (1.0f / NUM_ITER_F) - TRANSIT;
        float gt  = 1.0f / (1.0f + __expf(-EXP_BASE_F * tr));
        float eps = gt * (1.0f - H * (1.0f / LN_C));

        const float conf = 1.0f - ALPHA;
        const float base = ALPHA / (float)(CDIM - 1);
        // sum_j true_dist*(ln(true_dist) - p_j), using sum_j p_j == 1
        float cst   = conf * __logf(conf) + (float)(CDIM - 1) * base * __logf(base);
        float klsum = cst - conf * pt - base * (1.0f - pt);
        float klrow = klsum * (1.0f / (float)CDIM);

        ws[r * 3 + 0] = valid ? nll : 0.0f;
        ws[r * 3 + 1] = valid ? 1.0f : 0.0f;
        ws[r * 3 + 2] = klrow * eps;
    }
}

__global__ __launch_bounds__(256)
void finalize_kernel(const float* __restrict__ ws, float* __restrict__ out)
{
    __shared__ float rA[8], rB[8], rC[8];
    const int tid = threadIdx.x, lane = tid & 31, wave = tid >> 5;

    float a = 0.f, b = 0.f, c = 0.f;
    for (int i = tid; i < NROWS; i += 256) {
        a += ws[i * 3 + 0];
        b += ws[i * 3 + 1];
        c += ws[i * 3 + 2];
    }
    #pragma unroll
    for (int off = 16; off; off >>= 1) {
        a += __shfl_xor(a, off, 32);
        b += __shfl_xor(b, off, 32);
        c += __shfl_xor(c, off, 32);
    }
    if (lane == 0) { rA[wave] = a; rB[wave] = b; rC[wave] = c; }
    __syncthreads();
    if (tid == 0) {
        float A = 0.f, B = 0.f, C = 0.f;
        #pragma unroll
        for (int w = 0; w < 8; ++w) { A += rA[w]; B += rB[w]; C += rC[w]; }
        float nv = fmaxf(B, 1.0f);
        out[0] = A / nv + C * (1.0f / (float)NROWS);
    }
}

extern "C" void kernel_launch(void* const* d_in, const int* in_sizes, int n_in,
                              void* d_out, int out_size, void* d_ws, size_t ws_size,
                              hipStream_t stream) {
    const float* input    = (const float*)d_in[0];
    const int*   target   = (const int*)d_in[1];
    const int*   cur_time = (const int*)d_in[2];
    float*       ws       = (float*)d_ws;   // NROWS*3 floats, fully rewritten each call

    row_stats_kernel<<<NROWS, 256, 0, stream>>>(input, target, cur_time, ws);
    finalize_kernel<<<1, 256, 0, stream>>>(ws, (float*)d_out);
}